// SparseResBlock_42142219108964
// MI455X (gfx1250) — compile-verified
//
#include <hip/hip_runtime.h>

typedef __attribute__((ext_vector_type(2))) float v2f;
typedef __attribute__((ext_vector_type(8))) float v8f;

#define N_VOX 262144
#define CIN   16
#define COUT  32
#define KTAPS 27
#define EPSV  1e-5f

static __device__ __forceinline__ v8f wmma4(v2f a, v2f b, v8f c) {
  // V_WMMA_F32_16X16X4_F32 : D(16x16 f32) = A(16x4 f32) x B(4x16 f32) + C
  return __builtin_amdgcn_wmma_f32_16x16x4_f32(false, a, false, b, (short)0, c,
                                               false, false);
}

// WGP-scope prefetch (scope 0 = pull into ALL cache levels incl. WGP$).
// __builtin_prefetch only reaches SE/SYS scopes, which skip the WGP cache and
// only fill GL2 -- useless for an L2-resident gather stream.
static __device__ __forceinline__ void prefetch_wgp(const void* p) {
  asm volatile("global_prefetch_b8 %0, off" ::"v"(p));
}

// fast SiLU: v * rcp(1+exp(-v)) -> v_exp_f32 + v_rcp_f32 (TRANS ops, co-issue
// with WMMA) instead of the IEEE v_div_scale chain.
static __device__ __forceinline__ float silu_f(float v) {
  return v * __builtin_amdgcn_rcpf(1.0f + __expf(-v));
}

// ---------------------------------------------------------------------------
// prep: zero GN accumulators, build per-lane-swizzled B operands for WMMA.
// B layout per (tap k, kstep s, half nh, lane L):
//   col = 16*nh + (L%16), r0 = 4*s + 2*(L/16), value = {W[k][r0][col], W[k][r0+1][col]}
// ---------------------------------------------------------------------------
__global__ void prep_kernel(const float* __restrict__ W1, const float* __restrict__ W2,
                            const float* __restrict__ Wsk, float* __restrict__ stats,
                            v2f* __restrict__ wb1, v2f* __restrict__ wb2,
                            v2f* __restrict__ wbsk) {
  int t = blockIdx.x * blockDim.x + threadIdx.x;
  int stride = gridDim.x * blockDim.x;
  if (t < 96) stats[t] = 0.0f;  // sums1[16] sq1[16] sums2[32] sq2[32]
  for (int e = t; e < KTAPS * 4 * 2 * 32; e += stride) {
    int lane = e & 31; int r = e >> 5;
    int nh = r & 1; r >>= 1;
    int s = r & 3; int k = r >> 2;
    int col = 16 * nh + (lane & 15);
    int r0 = 4 * s + 2 * (lane >> 4);
    v2f w; w.x = W1[(k * CIN + r0) * COUT + col];
    w.y = W1[(k * CIN + r0 + 1) * COUT + col];
    wb1[e] = w;
  }
  for (int e = t; e < KTAPS * 8 * 2 * 32; e += stride) {
    int lane = e & 31; int r = e >> 5;
    int nh = r & 1; r >>= 1;
    int s = r & 7; int k = r >> 3;
    int col = 16 * nh + (lane & 15);
    int r0 = 4 * s + 2 * (lane >> 4);
    v2f w; w.x = W2[(k * COUT + r0) * COUT + col];
    w.y = W2[(k * COUT + r0 + 1) * COUT + col];
    wb2[e] = w;
  }
  for (int e = t; e < 4 * 2 * 32; e += stride) {
    int lane = e & 31; int r = e >> 5;
    int nh = r & 1; int s = r >> 1;
    int col = 16 * nh + (lane & 15);
    int r0 = 4 * s + 2 * (lane >> 4);
    v2f w; w.x = Wsk[r0 * COUT + col];
    w.y = Wsk[(r0 + 1) * COUT + col];
    wbsk[e] = w;
  }
}

// ---------------------------------------------------------------------------
// GroupNorm statistics: per-channel sum / sum-of-squares via LDS + f32 atomics
// ---------------------------------------------------------------------------
__global__ void stats_kernel(const float* __restrict__ src, float* __restrict__ sums,
                             float* __restrict__ sqs, int total, int cmask) {
  __shared__ float ls[32], lq[32];
  if (threadIdx.x < 32) { ls[threadIdx.x] = 0.0f; lq[threadIdx.x] = 0.0f; }
  __syncthreads();
  int t = blockIdx.x * blockDim.x + threadIdx.x;
  int stride = gridDim.x * blockDim.x;  // multiple of 32 -> channel invariant
  float s = 0.0f, q = 0.0f;
  for (int i = t; i < total; i += stride) { float v = src[i]; s += v; q += v * v; }
  int c = threadIdx.x & cmask;
  atomicAdd(&ls[c], s);
  atomicAdd(&lq[c], q);
  __syncthreads();
  if ((int)threadIdx.x <= cmask) {
    atomicAdd(&sums[threadIdx.x], ls[threadIdx.x]);
    atomicAdd(&sqs[threadIdx.x], lq[threadIdx.x]);
  }
}

// Fold GN into per-channel affine: h = x*scale[c] + shift[c]
__global__ void finalize_kernel(const float* __restrict__ sums, const float* __restrict__ sqs,
                                const float* __restrict__ gamma, const float* __restrict__ beta,
                                float* __restrict__ scale, float* __restrict__ shift,
                                int C, int cpg, float invCnt) {
  int c = threadIdx.x;
  if (c >= C) return;
  int g = c / cpg;
  float s = 0.0f, q = 0.0f;
  for (int u = 0; u < cpg; ++u) { s += sums[g * cpg + u]; q += sqs[g * cpg + u]; }
  float mean = s * invCnt;
  float var = q * invCnt - mean * mean;
  float inv = rsqrtf(var + EPSV);
  scale[c] = inv * gamma[c];
  shift[c] = beta[c] - mean * inv * gamma[c];
}

// ---------------------------------------------------------------------------
// conv1: GN1-on-the-fly gather + 27-tap WMMA GEMM (16->32) + SiLU -> a1
// Each wave32 owns TWO 16-row tiles (M=32): every B operand load feeds two
// v_wmma_f32_16x16x4_f32.  2-deep branchless index pipeline + WGP-scope
// prefetch of the row gathered one tap ahead.
// ---------------------------------------------------------------------------
__global__ __launch_bounds__(256) void conv1_kernel(
    const float* __restrict__ x, const int* __restrict__ nbr,
    const v2f* __restrict__ wb1, const float* __restrict__ scale1,
    const float* __restrict__ shift1, float* __restrict__ a1out) {
  __shared__ __align__(16) float lt[16 * 512];  // 16 tiles x (16x32) = 32 KB
  int lane = threadIdx.x & 31;
  int wave = threadIdx.x >> 5;
  int wtile = blockIdx.x * 8 + wave;
  int t0 = wtile * 2;
  int row = lane & 15;
  int hi = lane >> 4;
  int gr0 = t0 * 16 + row;
  int gr1 = gr0 + 16;

  v2f sc[4], sh[4];
#pragma unroll
  for (int s = 0; s < 4; ++s) {
    sc[s] = *(const v2f*)(scale1 + 4 * s + 2 * hi);
    sh[s] = *(const v2f*)(shift1 + 4 * s + 2 * hi);
  }
  v8f a00 = {}, a01 = {}, a10 = {}, a11 = {};
  const int* nb0 = nbr + (size_t)gr0 * KTAPS;
  const int* nb1 = nbr + (size_t)gr1 * KTAPS;
  int j0 = nb0[0], j1 = nb1[0];
  int jn0 = nb0[1], jn1 = nb1[1];
  prefetch_wgp(x + (size_t)j0 * CIN);
  prefetch_wgp(x + (size_t)j1 * CIN);
#pragma unroll 1
  for (int k = 0; k < KTAPS; ++k) {
    int kk = (k + 2 < KTAPS) ? (k + 2) : (KTAPS - 1);  // branchless clamp
    int jf0 = nb0[kk];
    int jf1 = nb1[kk];
    prefetch_wgp(x + (size_t)jn0 * CIN);   // rows needed one tap ahead
    prefetch_wgp(x + (size_t)jn1 * CIN);
    const float* xr0 = x + (size_t)j0 * CIN;
    const float* xr1 = x + (size_t)j1 * CIN;
#pragma unroll
    for (int s = 0; s < 4; ++s) {
      v2f a0 = *(const v2f*)(xr0 + 4 * s + 2 * hi);
      v2f a1 = *(const v2f*)(xr1 + 4 * s + 2 * hi);
      a0.x = a0.x * sc[s].x + sh[s].x;  a0.y = a0.y * sc[s].y + sh[s].y;
      a1.x = a1.x * sc[s].x + sh[s].x;  a1.y = a1.y * sc[s].y + sh[s].y;
      v2f b0 = wb1[((k * 4 + s) * 2 + 0) * 32 + lane];
      v2f b1 = wb1[((k * 4 + s) * 2 + 1) * 32 + lane];
      a00 = wmma4(a0, b0, a00);  a01 = wmma4(a0, b1, a01);
      a10 = wmma4(a1, b0, a10);  a11 = wmma4(a1, b1, a11);
    }
    j0 = jn0; j1 = jn1; jn0 = jf0; jn1 = jf1;
  }
  float* l0 = lt + (wave * 2 + 0) * 512;
  float* l1 = lt + (wave * 2 + 1) * 512;
#pragma unroll
  for (int v = 0; v < 8; ++v) {
    int m = v + 8 * hi;                 // C layout: N = lane%16, lane/16 -> M+8
    l0[m * 32 + row]      = silu_f(a00[v]);
    l0[m * 32 + 16 + row] = silu_f(a01[v]);
    l1[m * 32 + row]      = silu_f(a10[v]);
    l1[m * 32 + 16 + row] = silu_f(a11[v]);
  }
  __syncthreads();
  // block covers 16 contiguous tiles -> one fully coalesced 32KB store
  float4* gout = (float4*)(a1out + (size_t)blockIdx.x * 16 * 512);
  const float4* l4 = (const float4*)lt;
#pragma unroll
  for (int i = 0; i < 8; ++i) gout[i * 256 + threadIdx.x] = l4[i * 256 + threadIdx.x];
}

// ---------------------------------------------------------------------------
// conv2: GN2-on-the-fly gather + 27-tap WMMA GEMM (32->32) + SiLU
//        + WMMA skip (x @ Wsk + bsk, bias preloaded into accumulator) -> out
// Two 16-row tiles per wave, same scheme as conv1.
// ---------------------------------------------------------------------------
__global__ __launch_bounds__(256) void conv2_kernel(
    const float* __restrict__ a1, const int* __restrict__ nbr,
    const v2f* __restrict__ wb2, const float* __restrict__ scale2,
    const float* __restrict__ shift2, const float* __restrict__ x,
    const v2f* __restrict__ wbsk, const float* __restrict__ bsk,
    float* __restrict__ out) {
  __shared__ __align__(16) float lt[16 * 512];
  int lane = threadIdx.x & 31;
  int wave = threadIdx.x >> 5;
  int wtile = blockIdx.x * 8 + wave;
  int t0 = wtile * 2;
  int row = lane & 15;
  int hi = lane >> 4;
  int gr0 = t0 * 16 + row;
  int gr1 = gr0 + 16;

  v2f sc[8], sh[8];
#pragma unroll
  for (int s = 0; s < 8; ++s) {
    sc[s] = *(const v2f*)(scale2 + 4 * s + 2 * hi);
    sh[s] = *(const v2f*)(shift2 + 4 * s + 2 * hi);
  }
  v8f c00 = {}, c01 = {}, c10 = {}, c11 = {};
  const int* nb0 = nbr + (size_t)gr0 * KTAPS;
  const int* nb1 = nbr + (size_t)gr1 * KTAPS;
  int j0 = nb0[0], j1 = nb1[0];
  int jn0 = nb0[1], jn1 = nb1[1];
  prefetch_wgp(a1 + (size_t)j0 * COUT);
  prefetch_wgp(a1 + (size_t)j1 * COUT);
#pragma unroll 1
  for (int k = 0; k < KTAPS; ++k) {
    int kk = (k + 2 < KTAPS) ? (k + 2) : (KTAPS - 1);
    int jf0 = nb0[kk];
    int jf1 = nb1[kk];
    prefetch_wgp(a1 + (size_t)jn0 * COUT);
    prefetch_wgp(a1 + (size_t)jn1 * COUT);
    const float* xr0 = a1 + (size_t)j0 * COUT;
    const float* xr1 = a1 + (size_t)j1 * COUT;
#pragma unroll
    for (int s = 0; s < 8; ++s) {
      v2f a0 = *(const v2f*)(xr0 + 4 * s + 2 * hi);
      v2f a1v = *(const v2f*)(xr1 + 4 * s + 2 * hi);
      a0.x = a0.x * sc[s].x + sh[s].x;   a0.y = a0.y * sc[s].y + sh[s].y;
      a1v.x = a1v.x * sc[s].x + sh[s].x; a1v.y = a1v.y * sc[s].y + sh[s].y;
      v2f b0 = wb2[((k * 8 + s) * 2 + 0) * 32 + lane];
      v2f b1 = wb2[((k * 8 + s) * 2 + 1) * 32 + lane];
      c00 = wmma4(a0, b0, c00);   c01 = wmma4(a0, b1, c01);
      c10 = wmma4(a1v, b0, c10);  c11 = wmma4(a1v, b1, c11);
    }
    j0 = jn0; j1 = jn1; jn0 = jf0; jn1 = jf1;
  }
  // skip path: accumulators start at bias (N index is fixed per lane)
  float bias0 = bsk[row], bias1 = bsk[16 + row];
  v8f s00 = {bias0, bias0, bias0, bias0, bias0, bias0, bias0, bias0};
  v8f s01 = {bias1, bias1, bias1, bias1, bias1, bias1, bias1, bias1};
  v8f s10 = s00, s11 = s01;
  const float* xrow0 = x + (size_t)gr0 * CIN;
  const float* xrow1 = x + (size_t)gr1 * CIN;
#pragma unroll
  for (int s = 0; s < 4; ++s) {
    v2f a0 = *(const v2f*)(xrow0 + 4 * s + 2 * hi);
    v2f a1v = *(const v2f*)(xrow1 + 4 * s + 2 * hi);
    v2f b0 = wbsk[(s * 2 + 0) * 32 + lane];
    v2f b1 = wbsk[(s * 2 + 1) * 32 + lane];
    s00 = wmma4(a0, b0, s00);   s01 = wmma4(a0, b1, s01);
    s10 = wmma4(a1v, b0, s10);  s11 = wmma4(a1v, b1, s11);
  }
  float* l0 = lt + (wave * 2 + 0) * 512;
  float* l1 = lt + (wave * 2 + 1) * 512;
#pragma unroll
  for (int v = 0; v < 8; ++v) {
    int m = v + 8 * hi;
    l0[m * 32 + row]      = silu_f(c00[v]) + s00[v];
    l0[m * 32 + 16 + row] = silu_f(c01[v]) + s01[v];
    l1[m * 32 + row]      = silu_f(c10[v]) + s10[v];
    l1[m * 32 + 16 + row] = silu_f(c11[v]) + s11[v];
  }
  __syncthreads();
  float4* gout = (float4*)(out + (size_t)blockIdx.x * 16 * 512);
  const float4* l4 = (const float4*)lt;
#pragma unroll
  for (int i = 0; i < 8; ++i) gout[i * 256 + threadIdx.x] = l4[i * 256 + threadIdx.x];
}

// ---------------------------------------------------------------------------
extern "C" void kernel_launch(void* const* d_in, const int* in_sizes, int n_in,
                              void* d_out, int out_size, void* d_ws, size_t ws_size,
                              hipStream_t stream) {
  const float* x   = (const float*)d_in[0];
  const int*   nbr = (const int*)d_in[1];
  const float* g1  = (const float*)d_in[2];
  const float* b1  = (const float*)d_in[3];
  const float* W1  = (const float*)d_in[4];
  const float* g2  = (const float*)d_in[5];
  const float* b2  = (const float*)d_in[6];
  const float* W2  = (const float*)d_in[7];
  const float* Wsk = (const float*)d_in[8];
  const float* bsk = (const float*)d_in[9];
  float* out = (float*)d_out;
  float* ws  = (float*)d_ws;

  // workspace layout (floats)
  float* a1     = ws;                                   // N*32
  float* stats  = ws + (size_t)N_VOX * COUT;
  float* sums1  = stats + 0;
  float* sq1    = stats + 16;
  float* sums2  = stats + 32;
  float* sq2    = stats + 64;
  float* scale1 = stats + 96;
  float* shift1 = stats + 112;
  float* scale2 = stats + 128;
  float* shift2 = stats + 160;
  v2f* wb1  = (v2f*)(stats + 192);
  v2f* wb2  = wb1 + KTAPS * 4 * 2 * 32;
  v2f* wbsk = wb2 + KTAPS * 8 * 2 * 32;

  prep_kernel<<<32, 256, 0, stream>>>(W1, W2, Wsk, stats, wb1, wb2, wbsk);
  stats_kernel<<<256, 256, 0, stream>>>(x, sums1, sq1, N_VOX * CIN, 15);
  finalize_kernel<<<1, 32, 0, stream>>>(sums1, sq1, g1, b1, scale1, shift1,
                                        16, 2, 1.0f / ((float)N_VOX * 2.0f));
  conv1_kernel<<<N_VOX / 16 / 16, 256, 0, stream>>>(x, nbr, wb1, scale1, shift1, a1);
  stats_kernel<<<256, 256, 0, stream>>>(a1, sums2, sq2, N_VOX * COUT, 31);
  finalize_kernel<<<1, 32, 0, stream>>>(sums2, sq2, g2, b2, scale2, shift2,
                                        32, 4, 1.0f / ((float)N_VOX * 4.0f));
  conv2_kernel<<<N_VOX / 16 / 16, 256, 0, stream>>>(a1, nbr, wb2, scale2, shift2,
                                                    x, wbsk, bsk, out);
}